// CHMM_9861244912408
// MI455X (gfx1250) — compile-verified
//
#include <hip/hip_runtime.h>
#include <hip/hip_bf16.h>
#include <math.h>

typedef __attribute__((ext_vector_type(2))) float v2f;
typedef __attribute__((ext_vector_type(8))) float v8f;

#define I_DIM 1048576
#define B_DIM 32
#define J_DIM 32
#define TILES 8  // 16-row tiles per wave -> 128 rows/wave

// ---------------------------------------------------------------------------
// Prep kernel (1 block, 32 lanes):
//   ll[b]  = finite(max_j cp[b,j]) else 0                     -> ws[0..31]
//   B-frag table in per-lane WMMA layout                      -> ws[32..1055]
// B layout for V_WMMA_F32_16X16X4_F32 (4x16): VGPR v, lanes 0-15 = K-row v,
// lanes 16-31 = K-row v+2, N = lane&15. bfrag[(n*8+kk)*32 + lane] = float2
// {cp_e[krow][n*16+m], cp_e[krow+1][n*16+m]}, krow = 4kk + 2*(lane>>4).
// ---------------------------------------------------------------------------
__global__ __launch_bounds__(32) void chmm_prep(const float* __restrict__ cp,
                                                float* __restrict__ ws) {
    __shared__ float scpe[B_DIM * J_DIM];
    const int b = threadIdx.x;  // 0..31

    float mx = -INFINITY;
#pragma unroll
    for (int j = 0; j < J_DIM; ++j) mx = fmaxf(mx, cp[b * J_DIM + j]);
    if (!__builtin_isfinite(mx)) mx = 0.0f;
    ws[b] = mx;
#pragma unroll
    for (int j = 0; j < J_DIM; ++j) {
        float e = __expf(cp[b * J_DIM + j] - mx);  // cp - max <= 0 -> e <= 1
        if (!__builtin_isfinite(e)) e = 0.0f;
        scpe[b * J_DIM + j] = e;
    }
    __syncthreads();

    const int m    = b & 15;
    const int half = b >> 4;
#pragma unroll
    for (int n = 0; n < 2; ++n) {
#pragma unroll
        for (int kk = 0; kk < 8; ++kk) {
            const int krow = 4 * kk + 2 * half;
            v2f v;
            v.x = scpe[krow * J_DIM + n * 16 + m];
            v.y = scpe[(krow + 1) * J_DIM + n * 16 + m];
            *(v2f*)(ws + B_DIM + ((n * 8 + kk) * 32 + b) * 2) = v;
        }
    }
}

// ---------------------------------------------------------------------------
// Main kernel: 256 threads = 8 waves. Each wave: load B frags + ll ONCE into
// registers (loop-invariant), then stream TILES 16x32 row-tiles:
//   ratio = pf * __expf(ll - pp), isfinite-scrubbed (matches reference's
//   nan_to_num/zero-where-pp_e==0: underflow->0 directly, overflow->inf->0).
//   Two 16-wide N tiles, K=32 -> 8 chained V_WMMA_F32_16X16X4_F32 each.
// A layout (16x4 f32): lane L<16 holds (M=L, K=0/1), lane L>=16 (M=L-16, K=2/3)
// -> per kk each lane NT-loads float2 at cols {4kk+2*(lane>>4), +1} of its row.
// Each 32-float row = exactly one 128B line, fully consumed -> 1x DRAM traffic.
// ---------------------------------------------------------------------------
__global__ __launch_bounds__(256) void chmm_main(const float* __restrict__ pf,
                                                 const float* __restrict__ pp,
                                                 const float* __restrict__ ws,
                                                 float* __restrict__ out) {
    const int lane   = threadIdx.x & 31;
    const int wave   = threadIdx.x >> 5;
    const int m      = lane & 15;
    const int half   = lane >> 4;
    const int colsel = half * 2;

    // Loop-invariant operands -> registers (one pass of cached loads).
    const float* __restrict__ bf = ws + B_DIM;
    v2f Bf0[8], Bf1[8], llv[8];
#pragma unroll
    for (int kk = 0; kk < 8; ++kk) {
        llv[kk] = *(const v2f*)(ws + 4 * kk + colsel);          // {ll[c], ll[c+1]}
        Bf0[kk] = *(const v2f*)(bf + (kk * 32 + lane) * 2);     // N tile 0
        Bf1[kk] = *(const v2f*)(bf + ((8 + kk) * 32 + lane) * 2);  // N tile 1
    }

    const size_t rowBase = ((size_t)blockIdx.x * 8 + wave) * (16 * TILES);

    for (int t = 0; t < TILES; ++t) {
        const size_t r = rowBase + t * 16 + m;
        const float* __restrict__ pfp = pf + r * B_DIM;
        const float* __restrict__ ppp = pp + r * B_DIM;

        v8f c0 = {0.f, 0.f, 0.f, 0.f, 0.f, 0.f, 0.f, 0.f};
        v8f c1 = c0;

#pragma unroll
        for (int kk = 0; kk < 8; ++kk) {
            const int col = 4 * kk + colsel;
            v2f pfv = __builtin_nontemporal_load((const v2f*)(pfp + col));
            v2f ppv = __builtin_nontemporal_load((const v2f*)(ppp + col));

            // ratio = pf * exp(ll - pp); single native v_exp + mul, then a
            // branchless finite-scrub (v_cmp_class + v_cndmask keeps EXEC=~0).
            float e0 = __expf(llv[kk].x - ppv.x);
            float e1 = __expf(llv[kk].y - ppv.y);
            float a0 = pfv.x * e0;
            float a1 = pfv.y * e1;
            a0 = __builtin_isfinite(a0) ? a0 : 0.0f;
            a1 = __builtin_isfinite(a1) ? a1 : 0.0f;

            v2f A;
            A.x = a0;
            A.y = a1;

            // 8 args: (neg_a, A, neg_b, B, c_mod, C, reuse_a, reuse_b)
            c0 = __builtin_amdgcn_wmma_f32_16x16x4_f32(false, A, false, Bf0[kk],
                                                       (short)0, c0, false, false);
            c1 = __builtin_amdgcn_wmma_f32_16x16x4_f32(false, A, false, Bf1[kk],
                                                       (short)0, c1, false, false);
        }

        // C/D layout: VGPR v, lanes 0-15 -> (M=v, N=lane), lanes 16-31 -> M=v+8.
        // Streaming output -> NT stores; each half-wave writes 64B contiguous.
#pragma unroll
        for (int v = 0; v < 8; ++v) {
            const size_t row = rowBase + t * 16 + v + half * 8;
            __builtin_nontemporal_store(c0[v], out + row * J_DIM + m);
            __builtin_nontemporal_store(c1[v], out + row * J_DIM + 16 + m);
        }
    }
}

extern "C" void kernel_launch(void* const* d_in, const int* in_sizes, int n_in,
                              void* d_out, int out_size, void* d_ws, size_t ws_size,
                              hipStream_t stream) {
    const float* pf = (const float*)d_in[0];  // [I, B]
    const float* pp = (const float*)d_in[1];  // [I, B]
    const float* cp = (const float*)d_in[2];  // [B, J]
    float* out = (float*)d_out;               // [I, J]
    float* ws  = (float*)d_ws;                // 1056 floats used

    chmm_prep<<<1, 32, 0, stream>>>(cp, ws);
    // 1024 blocks x 8 waves x 128 rows = 1,048,576 rows
    chmm_main<<<I_DIM / (128 * TILES), 256, 0, stream>>>(pf, pp, ws, out);
}